// GNN_First_Layer_20547123544614
// MI455X (gfx1250) — compile-verified
//
#include <hip/hip_runtime.h>

// GNN first layer for MI455X (gfx1250), wave32 + WMMA f32 16x16x4.
//
// out = relu( atoms@Wv + residues@Wr + mean_g(atoms,sn)@Wsr + mean_g(atoms,dn)@Wdr )
// using linearity: mean_gather(atoms@W) == mean_gather(atoms)@W, so everything
// collapses to one [16x60]@[60x64] per 16-node tile, done as 15 K-steps of
// V_WMMA_F32_16X16X4_F32 across 4 N-tiles.
//
// LDS weight layout is packed as Wp[kstep][col][4] so each lane's B-fragment
// pair (rows 4kk+c, 4kk+c+1 at fixed col) is one 8B-aligned ds_load_b64.

#define N_NODES 200000
#define A_DIM   12
#define R_DIM   23
#define F_DIM   64
#define K_NB    10
#define XDIM    60      // 12 atoms + 23 residues + 1 zero pad + 12 same-mean + 12 diff-mean
#define KSTEPS  (XDIM / 4)
#define NTILES  (N_NODES / 16)   // 12500, exact
#define WAVES_PER_BLOCK 8
#define BLOCK   256

typedef __attribute__((ext_vector_type(2))) float v2f;
typedef __attribute__((ext_vector_type(8))) float v8f;

__global__ __launch_bounds__(BLOCK)
void gnn_first_layer_kernel(
    const float* __restrict__ atoms0, const float* __restrict__ residues0,
    const int*   __restrict__ sn0,    const int*   __restrict__ dn0,
    float*       __restrict__ out0,
    const float* __restrict__ atoms1, const float* __restrict__ residues1,
    const int*   __restrict__ sn1,    const int*   __restrict__ dn1,
    float*       __restrict__ out1,
    const float* __restrict__ Wv, const float* __restrict__ Wr,
    const float* __restrict__ Wsr, const float* __restrict__ Wdr)
{
    // Packed: Wl[kk*256 + col*4 + (r&3)] == W_cat[r][col], kk = r>>2
    __shared__ float Wl[KSTEPS * F_DIM * 4];                 // 15360 B
    __shared__ float Xl[WAVES_PER_BLOCK][16][XDIM];          // 30720 B

    const int  tid = threadIdx.x;
    const bool p1  = (blockIdx.y != 0);
    const float* atoms    = p1 ? atoms1    : atoms0;
    const float* residues = p1 ? residues1 : residues0;
    const int*   sn       = p1 ? sn1       : sn0;
    const int*   dn       = p1 ? dn1       : dn0;
    float*       out      = p1 ? out1      : out0;

    // ---- Phase 0: build packed W_cat (rows: Wv 0-11, Wr 12-34, pad 35, Wsr 36-47, Wdr 48-59)
    for (int i = tid; i < XDIM * F_DIM; i += BLOCK) {
        const int r = i >> 6, c = i & 63;
        float w;
        if      (r < 12)  w = Wv [ r       * F_DIM + c];
        else if (r < 35)  w = Wr [(r - 12) * F_DIM + c];
        else if (r == 35) w = 0.0f;
        else if (r < 48)  w = Wsr[(r - 36) * F_DIM + c];
        else              w = Wdr[(r - 48) * F_DIM + c];
        Wl[(r >> 2) * (F_DIM * 4) + c * 4 + (r & 3)] = w;
    }
    __syncthreads();

    const int wave = tid >> 5;
    const int lane = tid & 31;
    const int m    = lane & 15;      // node within tile / matrix row M
    const int hi   = lane >> 4;      // half-wave selector
    const int tile = blockIdx.x * WAVES_PER_BLOCK + wave;
    const bool active = (tile < NTILES);

    // ---- Phase 1: masked-mean gather of raw atom rows + stage x into LDS
    if (active) {
        const int node = tile * 16 + m;
        // lanes 0-15: same-neighbors of node m; lanes 16-31: diff-neighbors of node m
        const int* nb = (hi ? dn : sn) + (size_t)node * K_NB;
        float gacc[A_DIM];
#pragma unroll
        for (int j = 0; j < A_DIM; ++j) gacc[j] = 0.0f;
        float cnt = 0.0f;
#pragma unroll
        for (int k = 0; k < K_NB; ++k) {
            const int idx = nb[k];
            if (idx > -1) {                      // -1 marks padding
                cnt += 1.0f;
                const float4* row = reinterpret_cast<const float4*>(atoms + (size_t)idx * A_DIM);
                const float4 r0 = row[0], r1 = row[1], r2 = row[2];
                gacc[0] += r0.x; gacc[1] += r0.y; gacc[2]  += r0.z; gacc[3]  += r0.w;
                gacc[4] += r1.x; gacc[5] += r1.y; gacc[6]  += r1.z; gacc[7]  += r1.w;
                gacc[8] += r2.x; gacc[9] += r2.y; gacc[10] += r2.z; gacc[11] += r2.w;
            }
        }
        const float inv = 1.0f / fmaxf(cnt, 1.0f);
        float* xrow = &Xl[wave][m][0];
        if (hi == 0) {
            // own atoms
            const float4* arow = reinterpret_cast<const float4*>(atoms + (size_t)node * A_DIM);
            const float4 a0 = arow[0], a1 = arow[1], a2 = arow[2];
            xrow[0] = a0.x; xrow[1] = a0.y; xrow[2]  = a0.z; xrow[3]  = a0.w;
            xrow[4] = a1.x; xrow[5] = a1.y; xrow[6]  = a1.z; xrow[7]  = a1.w;
            xrow[8] = a2.x; xrow[9] = a2.y; xrow[10] = a2.z; xrow[11] = a2.w;
            // own residues
            const float* rrow = residues + (size_t)node * R_DIM;
#pragma unroll
            for (int j = 0; j < R_DIM; ++j) xrow[12 + j] = rrow[j];
            xrow[35] = 0.0f;                       // pad row
            // same-neighbor mean
#pragma unroll
            for (int j = 0; j < A_DIM; ++j) xrow[36 + j] = gacc[j] * inv;
        } else {
            // diff-neighbor mean
#pragma unroll
            for (int j = 0; j < A_DIM; ++j) xrow[48 + j] = gacc[j] * inv;
        }
    }
    __syncthreads();

    // ---- Phase 2: [16x60]@[60x64] via V_WMMA_F32_16X16X4_F32, 15 K-steps x 4 N-tiles
    if (active) {
        v8f acc[4];
#pragma unroll
        for (int nt = 0; nt < 4; ++nt) acc[nt] = (v8f){};

        const float* xw   = &Xl[wave][m][0];
        const int    coff = hi * 2;   // A/B frag: VGPR0 holds K0 (lanes 0-15) / K2 (lanes 16-31)

#pragma unroll
        for (int kk = 0; kk < KSTEPS; ++kk) {
            // A pair: x[m][4kk+coff], x[m][4kk+coff+1] — contiguous, 8B aligned
            const v2f a = *reinterpret_cast<const v2f*>(&xw[4 * kk + coff]);
#pragma unroll
            for (int nt = 0; nt < 4; ++nt) {
                const int col = nt * 16 + m;
                // B pair: W[4kk+coff][col], W[4kk+coff+1][col] — adjacent in packed layout
                const v2f b = *reinterpret_cast<const v2f*>(
                    &Wl[kk * (F_DIM * 4) + col * 4 + coff]);
                acc[nt] = __builtin_amdgcn_wmma_f32_16x16x4_f32(
                    /*neg_a=*/false, a, /*neg_b=*/false, b,
                    /*c_mod=*/(short)0, acc[nt], /*reuse_a=*/false, /*reuse_b=*/false);
            }
        }

        // D layout: VGPR r -> row r (lanes 0-15) / row r+8 (lanes 16-31), col = lane%16 per N-tile
        const size_t base = (size_t)tile * 16;
#pragma unroll
        for (int nt = 0; nt < 4; ++nt) {
#pragma unroll
            for (int r = 0; r < 8; ++r) {
                const int row = r + hi * 8;
                out[(base + row) * F_DIM + nt * 16 + m] = fmaxf(acc[nt][r], 0.0f);
            }
        }
    }
}

extern "C" void kernel_launch(void* const* d_in, const int* in_sizes, int n_in,
                              void* d_out, int out_size, void* d_ws, size_t ws_size,
                              hipStream_t stream) {
    (void)in_sizes; (void)n_in; (void)d_ws; (void)ws_size; (void)out_size;

    const float* atoms0    = (const float*)d_in[0];
    const float* residues0 = (const float*)d_in[1];
    const int*   sn0       = (const int*)  d_in[2];
    const int*   dn0       = (const int*)  d_in[3];
    const float* atoms1    = (const float*)d_in[4];
    const float* residues1 = (const float*)d_in[5];
    const int*   sn1       = (const int*)  d_in[6];
    const int*   dn1       = (const int*)  d_in[7];
    const float* Wv        = (const float*)d_in[8];
    const float* Wr        = (const float*)d_in[9];
    const float* Wsr       = (const float*)d_in[10];
    const float* Wdr       = (const float*)d_in[11];

    float* out = (float*)d_out;
    const size_t NF = (size_t)N_NODES * F_DIM;   // 12,800,000
    const size_t NK = (size_t)N_NODES * K_NB;    //  2,000,000

    // d_out layout (flat, return order): [out0 | sn0 | dn0 | out1 | sn1 | dn1]
    float* out0 = out;
    float* out1 = out + NF + 2 * NK;

    dim3 grid((NTILES + WAVES_PER_BLOCK - 1) / WAVES_PER_BLOCK, 2);
    gnn_first_layer_kernel<<<grid, dim3(BLOCK), 0, stream>>>(
        atoms0, residues0, sn0, dn0, out0,
        atoms1, residues1, sn1, dn1, out1,
        Wv, Wr, Wsr, Wdr);

    // neighbor-index passthrough (bit-exact copies)
    hipMemcpyAsync(out + NF,              (void*)sn0, NK * sizeof(int), hipMemcpyDeviceToDevice, stream);
    hipMemcpyAsync(out + NF + NK,         (void*)dn0, NK * sizeof(int), hipMemcpyDeviceToDevice, stream);
    hipMemcpyAsync(out + 2 * NF + 2 * NK, (void*)sn1, NK * sizeof(int), hipMemcpyDeviceToDevice, stream);
    hipMemcpyAsync(out + 2 * NF + 3 * NK, (void*)dn1, NK * sizeof(int), hipMemcpyDeviceToDevice, stream);
}